// ServiceGAT_67224828117291
// MI455X (gfx1250) — compile-verified
//
#include <hip/hip_runtime.h>
#include <hip/hip_bf16.h>
#include <math.h>

// ---------------------------------------------------------------------------
// GAT 2-layer pipeline for MI455X (gfx1250, wave32).
// Dense node-feature GEMMs use V_WMMA_F32_16X16X4_F32 (full fp32 precision;
// workload is scatter-bound, not matrix-bound). Each wave computes a 16x64
// output strip: one A fragment feeds 4 back-to-back WMMAs per k-step.
// ---------------------------------------------------------------------------

typedef __attribute__((ext_vector_type(2))) float v2f;
typedef __attribute__((ext_vector_type(8))) float v8f;

#define TPB 256

static inline int gridFor(long n, int b) { return (int)((n + b - 1) / b); }

// ------------------------------ utility ------------------------------------

__global__ void k_fill(float* __restrict__ p, long n, float v) {
  long i = blockIdx.x * (long)blockDim.x + threadIdx.x;
  if (i < n) p[i] = v;
}

__device__ inline void atomicMaxF(float* addr, float v) {
  // ordered-int trick: works for mixed-sign floats (addr init'd to -inf)
  if (v >= 0.0f) atomicMax((int*)addr, __float_as_int(v));
  else           atomicMin((unsigned int*)addr, __float_as_uint(v));
}

// sum of edge_attr (for self-loop fill='mean')
__global__ void k_sum_eattr(const float* __restrict__ e, int E, float* __restrict__ sumout) {
  __shared__ float sh[TPB];
  long i = blockIdx.x * (long)blockDim.x + threadIdx.x;
  float v = 0.0f;
  for (long j = i; j < E; j += (long)gridDim.x * blockDim.x) v += e[j];
  sh[threadIdx.x] = v;
  __syncthreads();
  for (int o = TPB / 2; o > 0; o >>= 1) {
    if ((int)threadIdx.x < o) sh[threadIdx.x] += sh[threadIdx.x + o];
    __syncthreads();
  }
  if (threadIdx.x == 0) atomicAdd(sumout, sh[0]);
}

// consts[0]=sum(eattr) (already accumulated), consts[1]=mean, consts[2..5]=ce0, consts[6..9]=ce1
__global__ void k_consts(const float* __restrict__ We0, const float* __restrict__ ae0, int C0,
                         const float* __restrict__ We1, const float* __restrict__ ae1, int C1,
                         float* __restrict__ consts, int E) {
  int t = threadIdx.x;
  if (t < 4) {
    float s = 0.0f;
    for (int c = 0; c < C0; ++c) s += We0[t * C0 + c] * ae0[t * C0 + c];
    consts[2 + t] = s;
  } else if (t < 8) {
    int h = t - 4;
    float s = 0.0f;
    for (int c = 0; c < C1; ++c) s += We1[h * C1 + c] * ae1[h * C1 + c];
    consts[6 + h] = s;
  } else if (t == 8) {
    consts[1] = consts[0] / (float)E;
  }
}

// ------------------------------ input proj ---------------------------------

__global__ void k_input_proj(const float* __restrict__ x, const float* __restrict__ W,
                             const float* __restrict__ b, float* __restrict__ h,
                             int N, int Fin, int Hd) {
  long i = blockIdx.x * (long)blockDim.x + threadIdx.x;
  if (i >= (long)N * Hd) return;
  int n = (int)(i / Hd), j = (int)(i % Hd);
  float s = b[j];
  for (int k = 0; k < Fin; ++k) s += x[(long)n * Fin + k] * W[k * Hd + j];
  h[i] = s;
}

// ------------------------------ WMMA GEMM ----------------------------------
// C[M,Nc] = A[M,K] * B[K,Nc], fp32. One wave owns a 16x(16*NT) output strip.
// V_WMMA_F32_16X16X4_F32 layouts (ISA 7.12.2):
//   A 16x4: v0 = K=0 (lanes 0-15) / K=2 (lanes 16-31); v1 = K=1 / K=3; lanes carry M.
//   B 4x16: symmetric, lanes carry N.
//   C/D:    VGPR r -> M=r (lanes 0-15) / M=r+8 (lanes 16-31); lanes carry N.
// Out-of-range M rows: A base clamped to row 0 and fragment multiplied by 0
// (branch-free -> EXEC stays all-ones, as WMMA requires; loads stay clauseable).
template <int NT>
__global__ void k_gemm_wmma(const float* __restrict__ A, const float* __restrict__ B,
                            float* __restrict__ C, int M, int K, int Nc) {
  int wavesPerBlock = blockDim.x >> 5;
  int wid = blockIdx.x * wavesPerBlock + ((int)threadIdx.x >> 5);
  int groupsN = Nc / (16 * NT);
  int tilesM = (M + 15) >> 4;
  if (wid >= tilesM * groupsN) return;          // wave-uniform exit
  int tm = wid / groupsN, tg = wid % groupsN;
  int lane = threadIdx.x & 31;
  int half = lane >> 4;
  int l = lane & 15;
  int row = tm * 16 + l;                        // M coordinate for A lanes
  int col0 = tg * 16 * NT + l;                  // first N coordinate for B/C lanes
  float msk = (row < M) ? 1.0f : 0.0f;
  const float* ap = A + (size_t)((row < M) ? row : 0) * K + 2 * half;

  v8f acc[NT];
#pragma unroll
  for (int t = 0; t < NT; ++t) acc[t] = (v8f){};

  for (int k = 0; k < K; k += 4) {
    if (k + 4 < K) {                            // wave-uniform, speculative prefetch
      __builtin_prefetch(ap + k + 4, 0, 1);
      __builtin_prefetch(B + (size_t)(k + 4 + 2 * half) * Nc + col0, 0, 1);
    }
    v2f a;
    a.x = ap[k] * msk;
    a.y = ap[k + 1] * msk;
    const float* bp = B + (size_t)(k + 2 * half) * Nc + col0;
#pragma unroll
    for (int t = 0; t < NT; ++t) {
      v2f bv;
      bv.x = bp[t * 16];
      bv.y = bp[Nc + t * 16];
      acc[t] = __builtin_amdgcn_wmma_f32_16x16x4_f32(
          /*neg_a=*/false, a, /*neg_b=*/false, bv,
          /*c_mod=*/(short)0, acc[t], /*reuse_a=*/false, /*reuse_b=*/false);
    }
  }
#pragma unroll
  for (int r = 0; r < 8; ++r) {
    int orow = tm * 16 + r + half * 8;
    if (orow < M) {
      float* cp = C + (size_t)orow * Nc + col0;
#pragma unroll
      for (int t = 0; t < NT; ++t) cp[t * 16] = acc[t][r];
    }
  }
}

// ------------------------------ attention ----------------------------------

// per-node, per-head dot products with a_s / a_d
__global__ void k_node_alpha(const float* __restrict__ x, const float* __restrict__ as,
                             const float* __restrict__ ad, float* __restrict__ asrc,
                             float* __restrict__ adst, int N, int C) {
  long i = blockIdx.x * (long)blockDim.x + threadIdx.x;
  if (i >= (long)N * 4) return;
  int n = (int)(i >> 2), h = (int)(i & 3);
  const float* xp = x + (long)n * 4 * C + h * C;
  const float* asp = as + h * C;
  const float* adp = ad + h * C;
  float s1 = 0.0f, s2 = 0.0f;
  for (int c = 0; c < C; ++c) { float v = xp[c]; s1 += v * asp[c]; s2 += v * adp[c]; }
  asrc[i] = s1;
  adst[i] = s2;
}

// alpha = leaky_relu(asrc[src]+adst[dst]+eattr*ce[h]); segment max via atomics
__global__ void k_edge_alpha_max(const float* __restrict__ asrc, const float* __restrict__ adst,
                                 const float* __restrict__ eattr, const int* __restrict__ src,
                                 const int* __restrict__ dst, const float* __restrict__ consts,
                                 int ceoff, float* __restrict__ alpha, float* __restrict__ m,
                                 int E, int N) {
  long i = blockIdx.x * (long)blockDim.x + threadIdx.x;
  long tot = (long)(E + N) * 4;
  if (i >= tot) return;
  long e = i >> 2; int h = (int)(i & 3);
  int s, d; float ea;
  if (e < E) { s = src[e]; d = dst[e]; ea = eattr[e]; }
  else       { s = d = (int)(e - E); ea = consts[1]; }
  float a = asrc[(long)s * 4 + h] + adst[(long)d * 4 + h] + ea * consts[ceoff + h];
  a = (a > 0.0f) ? a : 0.2f * a;   // leaky_relu, slope 0.2
  alpha[i] = a;
  atomicMaxF(&m[(long)d * 4 + h], a);
}

__global__ void k_edge_exp_sum(float* __restrict__ alpha, const float* __restrict__ m,
                               float* __restrict__ s, const int* __restrict__ dst,
                               int E, int N) {
  long i = blockIdx.x * (long)blockDim.x + threadIdx.x;
  long tot = (long)(E + N) * 4;
  if (i >= tot) return;
  long e = i >> 2; int h = (int)(i & 3);
  int d = (e < E) ? dst[e] : (int)(e - E);
  float ex = expf(alpha[i] - m[(long)d * 4 + h]);
  alpha[i] = ex;
  atomicAdd(&s[(long)d * 4 + h], ex);
}

__global__ void k_edge_norm(float* __restrict__ alpha, const float* __restrict__ s,
                            const int* __restrict__ dst, int E, int N) {
  long i = blockIdx.x * (long)blockDim.x + threadIdx.x;
  long tot = (long)(E + N) * 4;
  if (i >= tot) return;
  long e = i >> 2; int h = (int)(i & 3);
  int d = (e < E) ? dst[e] : (int)(e - E);
  alpha[i] = alpha[i] / (s[(long)d * 4 + h] + 1e-16f);
}

// out[dst] += alpha[e,h] * x[src]; float4 gather (a 4-span never crosses a head
// boundary since C is a multiple of 4 and spans are 4-aligned).
__global__ void k_aggregate(const float* __restrict__ x, const float* __restrict__ alpha,
                            const int* __restrict__ src, const int* __restrict__ dst,
                            float* __restrict__ out, int E, int N, int D, int C) {
  long i = blockIdx.x * (long)blockDim.x + threadIdx.x;
  int Dq = D >> 2;
  long tot = (long)(E + N) * Dq;
  if (i >= tot) return;
  long e = i / Dq;
  int r = (int)(i % Dq) << 2;
  int h = r / C;
  int sN, dN;
  if (e < E) { sN = src[e]; dN = dst[e]; }
  else       { sN = dN = (int)(e - E); }
  float w = alpha[e * 4 + h];
  const float4 xv = *(const float4*)(x + (long)sN * D + r);
  float* op = out + (long)dN * D + r;
  atomicAdd(op + 0, w * xv.x);
  atomicAdd(op + 1, w * xv.y);
  atomicAdd(op + 2, w * xv.z);
  atomicAdd(op + 3, w * xv.w);
}

// ------------------------------ layernorm ----------------------------------

__device__ inline float blockReduceSum(float v, float* sh, int D) {
  int lane = threadIdx.x & 31, wid = (int)threadIdx.x >> 5;
  for (int o = 16; o > 0; o >>= 1) v += __shfl_down(v, o, 32);
  if (lane == 0) sh[wid] = v;
  __syncthreads();
  int nw = D >> 5;
  if (wid == 0) {
    float t = (lane < nw) ? sh[lane] : 0.0f;
    for (int o = 16; o > 0; o >>= 1) t += __shfl_down(t, o, 32);
    if (lane == 0) sh[0] = t;
  }
  __syncthreads();
  float r = sh[0];
  __syncthreads();
  return r;
}

// one block per node, blockDim == D; bias add + layernorm + optional ELU
__global__ void k_bias_ln_act(const float* __restrict__ raw, const float* __restrict__ bias,
                              const float* __restrict__ g, const float* __restrict__ b,
                              float* __restrict__ out, int D, int do_elu) {
  __shared__ float sh[32];
  long n = blockIdx.x;
  int t = threadIdx.x;
  float v = raw[n * D + t] + bias[t];
  float mean = blockReduceSum(v, sh, D) / (float)D;
  float d = v - mean;
  float var = blockReduceSum(d * d, sh, D) / (float)D;
  float r = d * rsqrtf(var + 1e-5f) * g[t] + b[t];
  if (do_elu) r = (r > 0.0f) ? r : (expf(r) - 1.0f);
  out[n * D + t] = r;
}

// ------------------------------ pooling ------------------------------------

__global__ void k_pool_cnt(const int* __restrict__ batch, float* __restrict__ cnt, int N) {
  long i = blockIdx.x * (long)blockDim.x + threadIdx.x;
  if (i < N) atomicAdd(&cnt[batch[i]], 1.0f);
}

__global__ void k_pool_accum(const float* __restrict__ h, const int* __restrict__ batch,
                             float* __restrict__ psum, float* __restrict__ pmax, int N, int D) {
  long i = blockIdx.x * (long)blockDim.x + threadIdx.x;
  if (i >= (long)N * D) return;
  int n = (int)(i / D), f = (int)(i % D);
  int gI = batch[n];
  float v = h[i];
  atomicAdd(&psum[(long)gI * D + f], v);
  atomicMaxF(&pmax[(long)gI * D + f], v);
}

__global__ void k_pool_final(const float* __restrict__ psum, const float* __restrict__ pmax,
                             const float* __restrict__ cnt, float* __restrict__ out, int G, int D) {
  long i = blockIdx.x * (long)blockDim.x + threadIdx.x;
  if (i >= (long)G * D) return;
  int gI = (int)(i / D);
  float c = cnt[gI];
  if (c < 1.0f) c = 1.0f;
  out[i] = (psum[i] / c + pmax[i]) * 0.5f;
}

// ------------------------------ host driver --------------------------------

extern "C" void kernel_launch(void* const* d_in, const int* in_sizes, int n_in,
                              void* d_out, int out_size, void* d_ws, size_t ws_size,
                              hipStream_t stream) {
  const float* x      = (const float*)d_in[0];
  const int*   eidx   = (const int*)d_in[1];
  const float* eattr  = (const float*)d_in[2];
  const int*   batch  = (const int*)d_in[3];
  const float* W_in   = (const float*)d_in[4];
  const float* b_in   = (const float*)d_in[5];
  const float* l0_W   = (const float*)d_in[6];
  const float* l0_We  = (const float*)d_in[7];
  const float* l0_as  = (const float*)d_in[8];
  const float* l0_ad  = (const float*)d_in[9];
  const float* l0_ae  = (const float*)d_in[10];
  const float* l0_b   = (const float*)d_in[11];
  const float* ln0_g  = (const float*)d_in[12];
  const float* ln0_b  = (const float*)d_in[13];
  const float* l1_W   = (const float*)d_in[14];
  const float* l1_We  = (const float*)d_in[15];
  const float* l1_as  = (const float*)d_in[16];
  const float* l1_ad  = (const float*)d_in[17];
  const float* l1_ae  = (const float*)d_in[18];
  const float* l1_b   = (const float*)d_in[19];
  const float* ln1_g  = (const float*)d_in[20];
  const float* ln1_b  = (const float*)d_in[21];

  const int N   = in_sizes[3];          // 50000
  const int E   = in_sizes[2];          // 400000 (ed == 1)
  const int Fin = in_sizes[0] / N;      // 6
  const int Hd  = in_sizes[5];          // 64
  const int D0  = in_sizes[11];         // 256 = H * C0
  const int D1  = in_sizes[19];         // 128 = H * C1
  const int C0  = D0 / 4;
  const int C1  = D1 / 4;
  const int G   = 64;                   // NUM_GRAPHS
  const int* src = eidx;
  const int* dst = eidx + E;
  const long E2 = (long)E + N;

  // workspace bump allocator
  char* ws = (char*)d_ws;
  size_t off = 0;
  auto alloc = [&](size_t bytes) -> float* {
    float* p = (float*)(ws + off);
    off = (off + bytes + 255) & ~(size_t)255;
    return p;
  };
  float* h0     = alloc((size_t)N * Hd * 4);   // input-proj output
  float* xbuf   = alloc((size_t)N * D0 * 4);   // x0 -> act0
  float* aggbuf = alloc((size_t)N * D0 * 4);   // agg raw (reused layer1)
  float* x1     = alloc((size_t)N * D1 * 4);   // x1 -> act1
  float* alpha  = alloc((size_t)E2 * 4 * 4);
  float* asrc   = alloc((size_t)N * 4 * 4);
  float* adst   = alloc((size_t)N * 4 * 4);
  float* mbuf   = alloc((size_t)N * 4 * 4);
  float* sbuf   = alloc((size_t)N * 4 * 4);
  float* consts = alloc(16 * 4);
  float* psum   = alloc((size_t)G * D1 * 4);
  float* pmax   = alloc((size_t)G * D1 * 4);
  float* pcnt   = alloc((size_t)G * 4);

  // ---- constants: sum/mean of edge_attr, ce[h] for both layers ----
  k_fill<<<1, 64, 0, stream>>>(consts, 16, 0.0f);
  k_sum_eattr<<<256, TPB, 0, stream>>>(eattr, E, consts);
  k_consts<<<1, 64, 0, stream>>>(l0_We, l0_ae, C0, l1_We, l1_ae, C1, consts, E);

  // ---- input projection: h0 = x @ W_in + b_in ----
  k_input_proj<<<gridFor((long)N * Hd, TPB), TPB, 0, stream>>>(x, W_in, b_in, h0, N, Fin, Hd);

  // ---- layer 0 ----
  {
    int tiles = ((N + 15) / 16) * (D0 / 64);   // 16x64 strips per wave
    k_gemm_wmma<4><<<gridFor(tiles, 8), 256, 0, stream>>>(h0, l0_W, xbuf, N, Hd, D0);
    k_node_alpha<<<gridFor((long)N * 4, TPB), TPB, 0, stream>>>(xbuf, l0_as, l0_ad, asrc, adst, N, C0);
    k_fill<<<gridFor((long)N * 4, TPB), TPB, 0, stream>>>(mbuf, (long)N * 4, -INFINITY);
    k_fill<<<gridFor((long)N * 4, TPB), TPB, 0, stream>>>(sbuf, (long)N * 4, 0.0f);
    k_edge_alpha_max<<<gridFor(E2 * 4, TPB), TPB, 0, stream>>>(asrc, adst, eattr, src, dst,
                                                               consts, 2, alpha, mbuf, E, N);
    k_edge_exp_sum<<<gridFor(E2 * 4, TPB), TPB, 0, stream>>>(alpha, mbuf, sbuf, dst, E, N);
    k_edge_norm<<<gridFor(E2 * 4, TPB), TPB, 0, stream>>>(alpha, sbuf, dst, E, N);
    k_fill<<<gridFor((long)N * D0, TPB), TPB, 0, stream>>>(aggbuf, (long)N * D0, 0.0f);
    k_aggregate<<<gridFor(E2 * (D0 >> 2), TPB), TPB, 0, stream>>>(xbuf, alpha, src, dst,
                                                                  aggbuf, E, N, D0, C0);
    k_bias_ln_act<<<N, D0, 0, stream>>>(aggbuf, l0_b, ln0_g, ln0_b, xbuf, D0, /*elu=*/1);
  }

  // ---- layer 1 ----
  {
    int tiles = ((N + 15) / 16) * (D1 / 64);
    k_gemm_wmma<4><<<gridFor(tiles, 8), 256, 0, stream>>>(xbuf, l1_W, x1, N, D0, D1);
    k_node_alpha<<<gridFor((long)N * 4, TPB), TPB, 0, stream>>>(x1, l1_as, l1_ad, asrc, adst, N, C1);
    k_fill<<<gridFor((long)N * 4, TPB), TPB, 0, stream>>>(mbuf, (long)N * 4, -INFINITY);
    k_fill<<<gridFor((long)N * 4, TPB), TPB, 0, stream>>>(sbuf, (long)N * 4, 0.0f);
    k_edge_alpha_max<<<gridFor(E2 * 4, TPB), TPB, 0, stream>>>(asrc, adst, eattr, src, dst,
                                                               consts, 6, alpha, mbuf, E, N);
    k_edge_exp_sum<<<gridFor(E2 * 4, TPB), TPB, 0, stream>>>(alpha, mbuf, sbuf, dst, E, N);
    k_edge_norm<<<gridFor(E2 * 4, TPB), TPB, 0, stream>>>(alpha, sbuf, dst, E, N);
    k_fill<<<gridFor((long)N * D1, TPB), TPB, 0, stream>>>(aggbuf, (long)N * D1, 0.0f);
    k_aggregate<<<gridFor(E2 * (D1 >> 2), TPB), TPB, 0, stream>>>(x1, alpha, src, dst,
                                                                  aggbuf, E, N, D1, C1);
    k_bias_ln_act<<<N, D1, 0, stream>>>(aggbuf, l1_b, ln1_g, ln1_b, x1, D1, /*elu=*/0);
  }

  // ---- pooling: (mean + max) / 2 per graph ----
  k_fill<<<gridFor((long)G * D1, TPB), TPB, 0, stream>>>(psum, (long)G * D1, 0.0f);
  k_fill<<<gridFor((long)G * D1, TPB), TPB, 0, stream>>>(pmax, (long)G * D1, -INFINITY);
  k_fill<<<1, 64, 0, stream>>>(pcnt, G, 0.0f);
  k_pool_cnt<<<gridFor(N, TPB), TPB, 0, stream>>>(batch, pcnt, N);
  k_pool_accum<<<gridFor((long)N * D1, TPB), TPB, 0, stream>>>(x1, batch, psum, pmax, N, D1);
  k_pool_final<<<gridFor((long)G * D1, TPB), TPB, 0, stream>>>(psum, pmax, pcnt, (float*)d_out, G, D1);
}